// PKM_11836929868251
// MI455X (gfx1250) — compile-verified
//
#include <hip/hip_runtime.h>
#include <hip/hip_bf16.h>

// ---------------------------------------------------------------------------
// Product-key-memory attention for MI455X (gfx1250, wave32).
//
// Stage 1: Q = X(2048x512) @ Wq(512x1024)          -- V_WMMA_F32_16X16X4_F32
// Stage 2: LayerNorm(Q) in place
// Stage 3: dots[h,p] = Qhp(2048x128) @ Keys^T      -- V_WMMA_F32_16X16X4_F32
// Stage 4: two-stage top-32 + softmax (register-resident, mask-exclusion)
// Stage 5: out = sum_j attn[j] * values[idx[j]]    -- 512 MB gather, BW bound
// ---------------------------------------------------------------------------

typedef __attribute__((ext_vector_type(2))) float v2f;
typedef __attribute__((ext_vector_type(4))) float v4f;
typedef __attribute__((ext_vector_type(8))) float v8f;

#define ROWS   2048   // t*b
#define DIM    512    // embedding dim
#define DQ     1024   // dim_query = HEADS*DIM_HEAD
#define NHEAD  4
#define NK     512
#define KTOP   32
#define DHH    128    // DIM_HEAD/2

// ---------------------------------------------------------------------------
// Kernel 1: Q = X * Wq, fp32 WMMA 16x16x4.
// Each wave computes a 16x32 output slab (two 16x16 accumulators sharing the
// A fragment). Block = 8 waves => 16 rows x 256 cols per block.
// A frag: lanes 0-15 hold K={k,k+1} (v2f contiguous load), lanes 16-31 K={k+2,k+3}.
// B frag: v.x = B[k+2*half][n], v.y = B[k+2*half+1][n], n = n0 + (lane&15).
// ---------------------------------------------------------------------------
__global__ void __launch_bounds__(256) k_qproj(const float* __restrict__ X,
                                               const float* __restrict__ Wq,
                                               float* __restrict__ Q) {
  const int lane = threadIdx.x & 31;
  const int wave = threadIdx.x >> 5;
  const int half = lane >> 4;
  const int lm   = lane & 15;
  const int m0 = blockIdx.y * 16;
  const int n0 = blockIdx.x * 256 + wave * 32;

  const float* arow  = X  + (size_t)(m0 + lm) * DIM + half * 2;
  const float* bcol0 = Wq + (n0 + lm);

  v8f acc0 = {};
  v8f acc1 = {};
  for (int k = 0; k < DIM; k += 4) {
    v2f a = *(const v2f*)(arow + k);
    const int kr = (k + half * 2) * DQ;
    v2f b0, b1;
    b0.x = bcol0[kr];
    b0.y = bcol0[kr + DQ];
    b1.x = bcol0[kr + 16];
    b1.y = bcol0[kr + DQ + 16];
    acc0 = __builtin_amdgcn_wmma_f32_16x16x4_f32(false, a, false, b0, (short)0,
                                                 acc0, false, false);
    acc1 = __builtin_amdgcn_wmma_f32_16x16x4_f32(false, a, false, b1, (short)0,
                                                 acc1, false, false);
  }
  // C/D layout: VGPR i -> (M = i + 8*half, N = lane&15)
  #pragma unroll
  for (int i = 0; i < 8; ++i) {
    const int r = m0 + i + half * 8;
    Q[(size_t)r * DQ + n0 + lm]      = acc0[i];
    Q[(size_t)r * DQ + n0 + 16 + lm] = acc1[i];
  }
}

// ---------------------------------------------------------------------------
// Kernel 2: LayerNorm over each 1024-wide row of Q, in place.
// ---------------------------------------------------------------------------
__global__ void __launch_bounds__(256) k_lnorm(float* __restrict__ Q,
                                               const float* __restrict__ gamma,
                                               const float* __restrict__ beta) {
  const int r   = blockIdx.x;
  const int tid = threadIdx.x;
  float* row = Q + (size_t)r * DQ;

  v4f x = *(const v4f*)(row + tid * 4);
  float s  = x.x + x.y + x.z + x.w;
  float sq = x.x * x.x + x.y * x.y + x.z * x.z + x.w * x.w;
  #pragma unroll
  for (int o = 16; o >= 1; o >>= 1) {
    s  += __shfl_xor(s, o, 32);
    sq += __shfl_xor(sq, o, 32);
  }
  __shared__ float ss[8], ssq[8];
  const int lane = tid & 31, wave = tid >> 5;
  if (lane == 0) { ss[wave] = s; ssq[wave] = sq; }
  __syncthreads();
  float ts = 0.f, tq = 0.f;
  #pragma unroll
  for (int i = 0; i < 8; ++i) { ts += ss[i]; tq += ssq[i]; }

  const float mu  = ts * (1.0f / DQ);
  const float var = tq * (1.0f / DQ) - mu * mu;
  const float rs  = rsqrtf(var + 1e-6f);

  v4f g = *(const v4f*)(gamma + tid * 4);
  v4f b = *(const v4f*)(beta  + tid * 4);
  v4f y;
  y.x = (x.x - mu) * rs * g.x + b.x;
  y.y = (x.y - mu) * rs * g.y + b.y;
  y.z = (x.z - mu) * rs * g.z + b.z;
  y.w = (x.w - mu) * rs * g.w + b.w;
  *(v4f*)(row + tid * 4) = y;
}

// ---------------------------------------------------------------------------
// Kernel 3: dots[row,h,p,n] = sum_d Q[row, p*512 + h*128 + d] * Keys[h,n,p,d]
// Block = 8 waves, wave w = (h = w>>1, p = w&1); each wave does a 16x32 slab.
// Keys rows are contiguous in d, so the B fragment is a v2f contiguous load.
// ---------------------------------------------------------------------------
__global__ void __launch_bounds__(256) k_dots(const float* __restrict__ Q,
                                              const float* __restrict__ Keys,
                                              float* __restrict__ dots) {
  const int lane = threadIdx.x & 31;
  const int wave = threadIdx.x >> 5;
  const int h = wave >> 1;
  const int p = wave & 1;
  const int half = lane >> 4;
  const int lm   = lane & 15;
  const int m0 = blockIdx.y * 16;
  const int n0 = blockIdx.x * 32;

  const float* arow  = Q + (size_t)(m0 + lm) * DQ + p * (DQ / 2) + h * DHH + half * 2;
  const float* b0row = Keys + ((size_t)((h * NK + n0 + lm) * 2 + p)) * DHH + half * 2;
  const float* b1row = b0row + (size_t)16 * 2 * DHH;   // key n+16

  v8f acc0 = {};
  v8f acc1 = {};
  for (int k = 0; k < DHH; k += 4) {
    v2f a  = *(const v2f*)(arow + k);
    v2f b0 = *(const v2f*)(b0row + k);
    v2f b1 = *(const v2f*)(b1row + k);
    acc0 = __builtin_amdgcn_wmma_f32_16x16x4_f32(false, a, false, b0, (short)0,
                                                 acc0, false, false);
    acc1 = __builtin_amdgcn_wmma_f32_16x16x4_f32(false, a, false, b1, (short)0,
                                                 acc1, false, false);
  }
  #pragma unroll
  for (int i = 0; i < 8; ++i) {
    const int r = m0 + i + half * 8;
    const size_t base = (((size_t)r * NHEAD + h) * 2 + p) * NK;
    dots[base + n0 + lm]      = acc0[i];
    dots[base + n0 + 16 + lm] = acc1[i];
  }
}

// ---------------------------------------------------------------------------
// Kernel 4: per (row, head): top-32 of p0 row (512), top-32 of p1 row,
// combine 32x32, top-32 of 1024, softmax. One wave per (row, head).
// All candidate arrays stay in VGPRs: exclusion is tracked with bitmasks so
// no dynamically-indexed array writes occur (no scratch spills).
// ---------------------------------------------------------------------------
__global__ void __launch_bounds__(256) k_topk(const float* __restrict__ dots,
                                              float* __restrict__ attn,
                                              int* __restrict__ vidx) {
  const int lane = threadIdx.x & 31;
  const int wave = threadIdx.x >> 5;
  const int gw   = blockIdx.x * 8 + wave;
  const int row  = gw >> 2;
  const int h    = gw & 3;

  const float* d0 = dots + (((size_t)row * NHEAD + h) * 2 + 0) * NK;
  const float* d1 = d0 + NK;

  float v0[16], v1[16];
  #pragma unroll
  for (int j = 0; j < 16; ++j) {
    v0[j] = d0[lane + 32 * j];
    v1[j] = d1[lane + 32 * j];
  }

  const float NEG = -__builtin_inff();
  float s1 = NEG, s2 = NEG;
  int   i1 = 0,   i2 = 0;

  // ---- top-32 of v0 (element idx = lane + 32*j), winner #it lands in lane it
  unsigned used0 = 0u;
  #pragma unroll 1
  for (int it = 0; it < KTOP; ++it) {
    float bm = NEG; int bj = 0;
    #pragma unroll
    for (int j = 0; j < 16; ++j)
      if (!(used0 & (1u << j)) && v0[j] > bm) { bm = v0[j]; bj = j; }
    float rv = bm; int ri = lane + 32 * bj;
    #pragma unroll
    for (int o = 16; o >= 1; o >>= 1) {
      float ov = __shfl_xor(rv, o, 32);
      int   oi = __shfl_xor(ri, o, 32);
      if (ov > rv || (ov == rv && oi < ri)) { rv = ov; ri = oi; }
    }
    if (lane == it) { s1 = rv; i1 = ri; }
    if ((ri & 31) == lane) used0 |= 1u << (ri >> 5);
  }

  // ---- top-32 of v1
  unsigned used1 = 0u;
  #pragma unroll 1
  for (int it = 0; it < KTOP; ++it) {
    float bm = NEG; int bj = 0;
    #pragma unroll
    for (int j = 0; j < 16; ++j)
      if (!(used1 & (1u << j)) && v1[j] > bm) { bm = v1[j]; bj = j; }
    float rv = bm; int ri = lane + 32 * bj;
    #pragma unroll
    for (int o = 16; o >= 1; o >>= 1) {
      float ov = __shfl_xor(rv, o, 32);
      int   oi = __shfl_xor(ri, o, 32);
      if (ov > rv || (ov == rv && oi < ri)) { rv = ov; ri = oi; }
    }
    if (lane == it) { s2 = rv; i2 = ri; }
    if ((ri & 31) == lane) used1 |= 1u << (ri >> 5);
  }

  // ---- combine: c[j] = s1(lane) + s2(lane j); candidate gid = lane*32 + j
  float c[32];
  #pragma unroll
  for (int j = 0; j < 32; ++j) c[j] = s1 + __shfl(s2, j, 32);

  float fS = NEG; int fI = 0;
  unsigned used2 = 0u;
  #pragma unroll 1
  for (int it = 0; it < KTOP; ++it) {
    float bm = NEG; int bj = 0;
    #pragma unroll
    for (int j = 0; j < 32; ++j)
      if (!(used2 & (1u << j)) && c[j] > bm) { bm = c[j]; bj = j; }
    float rv = bm; int rg = (lane << 5) | bj;
    #pragma unroll
    for (int o = 16; o >= 1; o >>= 1) {
      float ov = __shfl_xor(rv, o, 32);
      int   og = __shfl_xor(rg, o, 32);
      if (ov > rv || (ov == rv && og < rg)) { rv = ov; rg = og; }
    }
    // combined value index of the winner (rg is uniform across the wave)
    const int wi1 = __shfl(i1, rg >> 5, 32);
    const int wi2 = __shfl(i2, rg & 31, 32);
    if (lane == it) { fS = rv; fI = wi1 * NK + wi2; }
    if ((rg >> 5) == lane) used2 |= 1u << (rg & 31);
  }

  // ---- softmax over the 32 winners (lane 0 holds the max by construction)
  const float mx = __shfl(fS, 0, 32);
  float e = __expf(fS - mx);
  float sum = e;
  #pragma unroll
  for (int o = 16; o >= 1; o >>= 1) sum += __shfl_xor(sum, o, 32);
  const float a = e / sum;

  attn[(size_t)row * 128 + h * 32 + lane] = a;
  vidx[(size_t)row * 128 + h * 32 + lane] = fI;
}

// ---------------------------------------------------------------------------
// Kernel 5: out[row] = sum_{j<128} w[j] * values[idx[j], :]   (512 cols)
// One block (256 threads) per token; each iteration streams one full 2 KB
// value row coalesced (float2 per lane). Prefetch indexed rows 4 iterations
// ahead (global_prefetch_b8) to hide index-dependent HBM latency.
// This stage moves ~512 MB total -> the whole problem's bandwidth floor.
// ---------------------------------------------------------------------------
__global__ void __launch_bounds__(256) k_gather(const float* __restrict__ values,
                                                const float* __restrict__ attn,
                                                const int* __restrict__ vidx,
                                                float* __restrict__ out) {
  const int row = blockIdx.x;
  const int tid = threadIdx.x;
  __shared__ float w[128];
  __shared__ int   id[128];
  if (tid < 128) {
    w[tid]  = attn[(size_t)row * 128 + tid];
    id[tid] = vidx[(size_t)row * 128 + tid];
  }
  __syncthreads();

  const int col = tid * 2;
  float ax = 0.f, ay = 0.f;
  #pragma unroll 4
  for (int j = 0; j < 128; ++j) {
    if (j + 4 < 128)
      __builtin_prefetch(values + (size_t)id[j + 4] * DIM + col, 0, 1);
    const v2f v = *(const v2f*)(values + (size_t)id[j] * DIM + col);
    ax += w[j] * v.x;
    ay += w[j] * v.y;
  }
  v2f o; o.x = ax; o.y = ay;
  *(v2f*)(out + (size_t)row * DIM + col) = o;
}

// ---------------------------------------------------------------------------
extern "C" void kernel_launch(void* const* d_in, const int* in_sizes, int n_in,
                              void* d_out, int out_size, void* d_ws, size_t ws_size,
                              hipStream_t stream) {
  const float* x        = (const float*)d_in[0];  // (1024, 2, 512)
  const float* Wq       = (const float*)d_in[1];  // (512, 1024)
  const float* ln_scale = (const float*)d_in[2];  // (1024,)
  const float* ln_bias  = (const float*)d_in[3];  // (1024,)
  const float* keys     = (const float*)d_in[4];  // (4, 512, 2, 128)
  const float* values   = (const float*)d_in[5];  // (262144, 512)
  float* out = (float*)d_out;                     // (1024, 2, 512)

  // Workspace carve-up (42 MB total)
  float* q    = (float*)d_ws;                         // 2048*1024
  float* dots = q + (size_t)ROWS * DQ;                // 2048*4*2*512
  float* attn = dots + (size_t)ROWS * NHEAD * 2 * NK; // 2048*128
  int*   vidx = (int*)(attn + (size_t)ROWS * 128);    // 2048*128

  k_qproj <<<dim3(DQ / 256, ROWS / 16), 256, 0, stream>>>(x, Wq, q);
  k_lnorm <<<ROWS,                      256, 0, stream>>>(q, ln_scale, ln_bias);
  k_dots  <<<dim3(NK / 32, ROWS / 16),  256, 0, stream>>>(q, keys, dots);
  k_topk  <<<ROWS * NHEAD / 8,          256, 0, stream>>>(dots, attn, vidx);
  k_gather<<<ROWS,                      256, 0, stream>>>(values, attn, vidx, out);
}